// GaussianSplatRenderer_46892452937925
// MI455X (gfx1250) — compile-verified
//
#include <hip/hip_runtime.h>
#include <math.h>

typedef __attribute__((ext_vector_type(2))) float v2f;
typedef __attribute__((ext_vector_type(4))) float f4v;
typedef __attribute__((ext_vector_type(8))) float v8f;
typedef int gv4i __attribute__((vector_size(16)));   // matches builtin param type

#define N_G   2048
#define W_IMG 128
#define H_IMG 128
#define FX_C  160.0f
#define FY_C  160.0f
#define CX_C  64.0f
#define CY_C  64.0f
#define SH_C0 0.28209479177387814f
#define COV_DIL 0.3f

#if defined(__has_builtin)
#if __has_builtin(__builtin_amdgcn_global_load_async_to_lds_b128) && \
    __has_builtin(__builtin_amdgcn_s_wait_asynccnt)
#define HAVE_ASYNC_LDS 1
#endif
#endif
#ifndef HAVE_ASYNC_LDS
#define HAVE_ASYNC_LDS 0
#endif

// ---------------------------------------------------------------------------
// Kernel 1: per-gaussian preprocessing.
// Produces: tz (sort key), 8 polynomial coefficients of power(gx,gy), and
// packed {opacity(masked), r, g, b}.
// power = c0 + c1*gx + c2*gy + c3*gx^2 + c4*gx*gy + c5*gy^2
// ---------------------------------------------------------------------------
__global__ __launch_bounds__(256) void gsplat_prep(
    const float* __restrict__ means, const float* __restrict__ harm,
    const float* __restrict__ opac,  const float* __restrict__ scales,
    const float* __restrict__ rots,  const float* __restrict__ vm,
    float* __restrict__ tzb, float* __restrict__ coefU, f4v* __restrict__ opcolU)
{
    int i = blockIdx.x * 256 + threadIdx.x;
    if (i >= N_G) return;

    // color from degree-0 SH
    float colr = fmaxf(SH_C0 * harm[3*i+0] + 0.5f, 0.f);
    float colg = fmaxf(SH_C0 * harm[3*i+1] + 0.5f, 0.f);
    float colb = fmaxf(SH_C0 * harm[3*i+2] + 0.5f, 0.f);

    // quaternion -> rotation
    float qw = rots[4*i+0], qx = rots[4*i+1], qy = rots[4*i+2], qz = rots[4*i+3];
    float qinv = rsqrtf(qw*qw + qx*qx + qy*qy + qz*qz);
    qw *= qinv; qx *= qinv; qy *= qinv; qz *= qinv;
    float R[3][3];
    R[0][0] = 1.f - 2.f*(qy*qy + qz*qz); R[0][1] = 2.f*(qx*qy - qw*qz); R[0][2] = 2.f*(qx*qz + qw*qy);
    R[1][0] = 2.f*(qx*qy + qw*qz); R[1][1] = 1.f - 2.f*(qx*qx + qz*qz); R[1][2] = 2.f*(qy*qz - qw*qx);
    R[2][0] = 2.f*(qx*qz - qw*qy); R[2][1] = 2.f*(qy*qz + qw*qx); R[2][2] = 1.f - 2.f*(qx*qx + qy*qy);

    float sc[3] = { scales[3*i+0], scales[3*i+1], scales[3*i+2] };
    float M[3][3];
#pragma unroll
    for (int r = 0; r < 3; ++r)
#pragma unroll
        for (int c = 0; c < 3; ++c) M[r][c] = R[r][c] * sc[c];

    float Sig[3][3];
#pragma unroll
    for (int r = 0; r < 3; ++r)
#pragma unroll
        for (int l = 0; l < 3; ++l)
            Sig[r][l] = M[r][0]*M[l][0] + M[r][1]*M[l][1] + M[r][2]*M[l][2];

    float Wm[3][3], pv[3];
#pragma unroll
    for (int r = 0; r < 3; ++r) {
#pragma unroll
        for (int c = 0; c < 3; ++c) Wm[r][c] = vm[4*r+c];
        pv[r] = vm[4*r+3];
    }

    // camera-space position
    float t0 = Wm[0][0]*means[3*i] + Wm[0][1]*means[3*i+1] + Wm[0][2]*means[3*i+2] + pv[0];
    float t1 = Wm[1][0]*means[3*i] + Wm[1][1]*means[3*i+1] + Wm[1][2]*means[3*i+2] + pv[1];
    float t2 = Wm[2][0]*means[3*i] + Wm[2][1]*means[3*i+1] + Wm[2][2]*means[3*i+2] + pv[2];
    float tz = fmaxf(t2, 0.01f);
    float op = (t2 > 0.2f) ? opac[i] : 0.f;     // invalid -> alpha 0

    float u = FX_C * t0 / tz + CX_C;
    float v = FY_C * t1 / tz + CY_C;

    // cov_cam = W Sigma W^T
    float T[3][3];
#pragma unroll
    for (int r = 0; r < 3; ++r)
#pragma unroll
        for (int k = 0; k < 3; ++k)
            T[r][k] = Wm[r][0]*Sig[0][k] + Wm[r][1]*Sig[1][k] + Wm[r][2]*Sig[2][k];
    float C[3][3];
#pragma unroll
    for (int r = 0; r < 3; ++r)
#pragma unroll
        for (int l = 0; l < 3; ++l)
            C[r][l] = T[r][0]*Wm[l][0] + T[r][1]*Wm[l][1] + T[r][2]*Wm[l][2];

    // J C J^T  with sparse 2x3 Jacobian
    float J00 = FX_C / tz, J02 = -FX_C * t0 / (tz * tz);
    float J11 = FY_C / tz, J12 = -FY_C * t1 / (tz * tz);
    float JC0[3], JC1[3];
#pragma unroll
    for (int k = 0; k < 3; ++k) {
        JC0[k] = J00 * C[0][k] + J02 * C[2][k];
        JC1[k] = J11 * C[1][k] + J12 * C[2][k];
    }
    float a = JC0[0]*J00 + JC0[2]*J02 + COV_DIL;
    float b = JC0[1]*J11 + JC0[2]*J12;
    float c = JC1[1]*J11 + JC1[2]*J12 + COV_DIL;

    float det = a*c - b*b;
    float ia =  c / det;
    float ib = -b / det;
    float ic =  a / det;

    // quadratic polynomial coefficients of power(gx,gy)
    float c0 = -0.5f*(ia*u*u + ic*v*v) - ib*u*v;
    float c1 = ia*u + ib*v;
    float c2 = ic*v + ib*u;
    float c3 = -0.5f*ia;
    float c4 = -ib;
    float c5 = -0.5f*ic;

    coefU[i*8+0] = c0; coefU[i*8+1] = c1; coefU[i*8+2] = c2; coefU[i*8+3] = c3;
    coefU[i*8+4] = c4; coefU[i*8+5] = c5; coefU[i*8+6] = 0.f; coefU[i*8+7] = 0.f;
    f4v oc; oc.x = op; oc.y = colr; oc.z = colg; oc.w = colb;
    opcolU[i] = oc;
    tzb[i] = tz;
}

// ---------------------------------------------------------------------------
// Kernel 2: stable rank-by-counting sort (equivalent to argsort(tz)) and
// scatter of coefficient rows + op/color into depth order. One block.
// ---------------------------------------------------------------------------
__global__ __launch_bounds__(1024) void gsplat_sort(
    const float* __restrict__ tzb, const float* __restrict__ coefU,
    const f4v* __restrict__ opcolU, float* __restrict__ coefS,
    f4v* __restrict__ opcolS)
{
    __shared__ float keys[N_G];
    int tid = threadIdx.x;
    keys[tid]        = tzb[tid];
    keys[tid + 1024] = tzb[tid + 1024];
    __syncthreads();

    for (int i = tid; i < N_G; i += 1024) {
        float k = keys[i];
        int rank = 0;
        for (int j = 0; j < N_G; ++j) {
            float kj = keys[j];                  // LDS broadcast read
            rank += (kj < k) || (kj == k && j < i);
        }
#pragma unroll
        for (int q = 0; q < 8; ++q) coefS[rank*8 + q] = coefU[i*8 + q];
        opcolS[rank] = opcolU[i];
    }
}

// ---------------------------------------------------------------------------
// Kernel 3: render. Wave32: each wave owns 16 pixels, replicated in the two
// lane halves. Per 16-gaussian chunk:
//   power tile (16 gaussians x 16 pixels) via 2x V_WMMA_F32_16X16X4_F32
//   (M = gaussians in VGPR rows, N = pixels in lanes, K = 8 poly features).
// Transmittance scan: serial over the 8 VGPR rows per lane; cross-half prefix
// via __shfl_xor(16); final color reduce via __shfl_xor(16).
// {op,color} table staged via async global->LDS (ASYNCcnt) when available.
// ---------------------------------------------------------------------------
__global__ __launch_bounds__(256) void gsplat_render(
    const float* __restrict__ coefS, const f4v* __restrict__ opcolS,
    float* __restrict__ out)
{
    __shared__ f4v oc_s[N_G];                    // 32 KB: {op, r, g, b} sorted
    const int tid = threadIdx.x;
#if HAVE_ASYNC_LDS
    for (int i = tid; i < N_G; i += 256) {
        __builtin_amdgcn_global_load_async_to_lds_b128(
            (__attribute__((address_space(1))) gv4i*)(void*)(opcolS + i),
            (__attribute__((address_space(3))) gv4i*)(void*)(&oc_s[i]),
            /*imm offset*/0, /*cpol*/0);
    }
    __builtin_amdgcn_s_wait_asynccnt(0);
#else
    for (int i = tid; i < N_G; i += 256) oc_s[i] = opcolS[i];
#endif
    __syncthreads();

    const int lane = tid & 31;
    const int hi   = lane >> 4;                  // which 8-gaussian half
    const int n    = lane & 15;                  // pixel slot within wave tile
    const int wave = tid >> 5;
    const int px   = blockIdx.x * 128 + wave * 16 + n;   // flat pixel id
    const float gx = (float)(px & (W_IMG - 1)) + 0.5f;
    const float gy = (float)(px >> 7) + 0.5f;

    // B matrix (K x N = 8 x 16 pixel features), constant across all chunks.
    // VGPR0 of half hi carries K=2*hi (+4 for second WMMA), VGPR1 carries +1.
    float f[8] = { 1.f, gx, gy, gx*gx, gx*gy, gy*gy, 0.f, 0.f };
    v2f b0, b1;
    b0[0] = f[2*hi + 0]; b0[1] = f[2*hi + 1];
    b1[0] = f[2*hi + 4]; b1[1] = f[2*hi + 5];

    const v2f* arow = (const v2f*)coefS;         // gaussian row = 4 v2f

    float Trun = 1.f;                            // running transmittance
    float cr = 0.f, cg = 0.f, cb = 0.f;          // partial color (this half)

    for (int base = 0; base < N_G; base += 16) {
        if (base + 16 < N_G)
            __builtin_prefetch(&coefS[(base + 16) * 8], 0, 0);

        // A matrix: 16 gaussians x 8 coefs (two K=4 slabs)
        int gi = base + n;
        v2f a0 = arow[gi*4 + hi];                // coefs[gi][2*hi .. 2*hi+1]
        v2f a1 = arow[gi*4 + 2 + hi];            // coefs[gi][4+2*hi .. 5+2*hi]

        v8f acc = {};
        acc = __builtin_amdgcn_wmma_f32_16x16x4_f32(
                false, a0, false, b0, (short)0, acc, false, false);
        acc = __builtin_amdgcn_wmma_f32_16x16x4_f32(
                false, a1, false, b1, (short)0, acc, false, false);

        // alpha for the 8 gaussians this lane holds (sorted order)
        const int gbase = base + hi * 8;
        float al[8];
#pragma unroll
        for (int r = 0; r < 8; ++r) {
            float op = oc_s[gbase + r].x;        // LDS broadcast within half
            float pw = fminf(acc[r], 0.f);
            al[r] = fminf(op * __expf(pw), 0.99f);
        }

        // serial in-lane transmittance scan over this half's 8 gaussians
        float Tself = 1.f, w[8];
#pragma unroll
        for (int r = 0; r < 8; ++r) {
            w[r]   = Tself * al[r];
            Tself *= (1.f - al[r]);
        }

        // cross-half prefix: hi half starts after lo half's 8 gaussians
        float Pother = __shfl_xor(Tself, 16, 32);
        float Tstart = Trun * (hi ? Pother : 1.f);

#pragma unroll
        for (int r = 0; r < 8; ++r) {
            f4v oc = oc_s[gbase + r];
            float ww = Tstart * w[r];
            cr += ww * oc.y;
            cg += ww * oc.z;
            cb += ww * oc.w;
        }
        Trun *= Tself * Pother;                  // both halves advance by 16
    }

    // combine the two half-contributions for this pixel
    cr += __shfl_xor(cr, 16, 32);
    cg += __shfl_xor(cg, 16, 32);
    cb += __shfl_xor(cb, 16, 32);
    if (!hi) {
        out[px*3 + 0] = cr;
        out[px*3 + 1] = cg;
        out[px*3 + 2] = cb;
    }
}

// ---------------------------------------------------------------------------
extern "C" void kernel_launch(void* const* d_in, const int* in_sizes, int n_in,
                              void* d_out, int out_size, void* d_ws, size_t ws_size,
                              hipStream_t stream) {
    const float* means  = (const float*)d_in[0];
    const float* harm   = (const float*)d_in[1];
    const float* opac   = (const float*)d_in[2];
    const float* scales = (const float*)d_in[3];
    const float* rots   = (const float*)d_in[4];
    const float* vm     = (const float*)d_in[5];
    float* out = (float*)d_out;

    float* ws     = (float*)d_ws;
    float* tzb    = ws;                                   // 2048 f
    float* coefU  = ws + 2048;                            // 16384 f
    f4v*   opcolU = (f4v*)(ws + 2048 + 16384);            // 2048 f4 (16B aligned)
    float* coefS  = ws + 2048 + 16384 + 8192;             // 16384 f
    f4v*   opcolS = (f4v*)(ws + 2048 + 16384 + 8192 + 16384);

    gsplat_prep<<<N_G / 256, 256, 0, stream>>>(means, harm, opac, scales, rots, vm,
                                               tzb, coefU, opcolU);
    gsplat_sort<<<1, 1024, 0, stream>>>(tzb, coefU, opcolU, coefS, opcolS);
    gsplat_render<<<(H_IMG * W_IMG) / 128, 256, 0, stream>>>(coefS, opcolS, out);
}